// BilateralGrid_20942260536131
// MI455X (gfx1250) — compile-verified
//
#include <hip/hip_runtime.h>
#include <stdint.h>

typedef __attribute__((ext_vector_type(4))) unsigned int v4u;
typedef __attribute__((ext_vector_type(8))) unsigned int v8u;

#define GC 12
#define GL 8
#define GH 16
#define GW 16
#define GCELLS (GL * GH * GW)            /* 2048 grid cells (corners)   */
#define GRID_ELEMS (GC * GCELLS)         /* 24576 floats                */
#define GRID_BYTES (GRID_ELEMS * 4)      /* 98304 bytes = 96 KB         */

// ---------------------------------------------------------------------------
// TDM: stage one image's 96 KB grid (channel-major) into LDS region A.
// D# per CDNA5 ISA; 2-group form (tensor <= 2D).
// ---------------------------------------------------------------------------
__device__ __forceinline__ void tdm_load_grid_to_lds(const float* gsrc,
                                                     uint32_t lds_off) {
  uint64_t ga = (uint64_t)(uintptr_t)gsrc;

  v4u g0;
  g0.x = 1u;                                   // count=1, is_restore=0
  g0.y = lds_off;                              // lds_addr (bytes)
  g0.z = (uint32_t)ga;                         // global_addr[31:0]
  g0.w = (uint32_t)((ga >> 32) & 0x01FFFFFFull) | (2u << 30); // [56:32]|type=2

  const uint32_t TD0   = GRID_ELEMS;           // tensor_dim0 (elements)
  const uint32_t TD1   = 1u;                   // tensor_dim1
  const uint32_t TILE0 = GRID_ELEMS;           // tile_dim0 (fits 16 bits)
  const uint32_t TILE1 = 1u;                   // tile_dim1
  const uint64_t STR0  = GRID_ELEMS;           // tensor_dim0_stride

  v8u g1;
  g1[0] = (2u << 16);                          // data_size=2 (4B), no multicast
  g1[1] = (TD0 & 0xFFFFu) << 16;               // tensor_dim0[15:0] @ bits 63:48
  g1[2] = (TD0 >> 16) | ((TD1 & 0xFFFFu) << 16);
  g1[3] = (TD1 >> 16) | ((TILE0 & 0xFFFFu) << 16);
  g1[4] = (TILE1 & 0xFFFFu);                   // tile_dim1 | tile_dim2=0
  g1[5] = (uint32_t)(STR0 & 0xFFFFFFFFull);    // stride0[31:0]
  g1[6] = (uint32_t)(STR0 >> 32);              // stride0[47:32]
  g1[7] = 0u;

  asm volatile("tensor_load_to_lds %0, %1" : : "s"(g0), "s"(g1) : "memory");
}

// ---------------------------------------------------------------------------
// 1024-thread block (32 waves, one WGP): TDM-stage channel-major grid into
// LDS region A, transpose to corner-major LDS region B (cell*12 + ch, 48B stride => every
// cell base 16B aligned), then grid-stride pixel loop with 3x ds_load_b128
// per corner.
// ---------------------------------------------------------------------------
__global__ void __launch_bounds__(1024)
bgrid_slice_kernel(const float* __restrict__ grid_xy,
                   const float* __restrict__ rgb,
                   const float* __restrict__ grids,
                   float* __restrict__ out,
                   int P /* pixels per image */) {
  extern __shared__ float smem[];          // [0,24576) = A, [24576,49152) = B
  float* __restrict__ Ag = smem;
  float* __restrict__ Bg = smem + GRID_ELEMS;
  const int n = blockIdx.y;

  if (threadIdx.x == 0) {                  // wave 0 issues the DMA once
    tdm_load_grid_to_lds(grids + (size_t)n * GRID_ELEMS,
                         (uint32_t)(uintptr_t)(void*)Ag);
    __builtin_amdgcn_s_wait_tensorcnt(0);
  }
  __syncthreads();

  // One-time transpose: A[ch*2048 + cell] -> B[cell*12 + ch]
  for (int e = threadIdx.x; e < GRID_ELEMS; e += blockDim.x) {
    int ch = e >> 11;                      // e / 2048
    int c  = e & (GCELLS - 1);             // e % 2048
    Bg[c * GC + ch] = Ag[e];
  }
  __syncthreads();

  const float2* __restrict__ xy2 = (const float2*)grid_xy + (size_t)n * P;
  const float*  __restrict__ cpx = rgb + (size_t)n * P * 3;
  float*        __restrict__ opx = out + (size_t)n * P * 3;

  const int stride = gridDim.x * blockDim.x;
  for (int p = blockIdx.x * blockDim.x + threadIdx.x; p < P; p += stride) {
    float2 xy = xy2[p];                                    // global_load_b64
    float r = cpx[3 * p + 0];
    float g = cpx[3 * p + 1];
    float b = cpx[3 * p + 2];
    float z = fmaf(0.299f, r, fmaf(0.587f, g, 0.114f * b));

    // align_corners=True sampling coords
    float ix = xy.x * (float)(GW - 1);
    float iy = xy.y * (float)(GH - 1);
    float iz = z    * (float)(GL - 1);
    float xf = floorf(ix), yf = floorf(iy), zf = floorf(iz);
    float wx = ix - xf, wy = iy - yf, wz = iz - zf;

    int x0 = min(max((int)xf, 0), GW - 1);
    int y0 = min(max((int)yf, 0), GH - 1);
    int z0 = min(max((int)zf, 0), GL - 1);
    int x1 = min(x0 + 1, GW - 1);
    int y1 = min(y0 + 1, GH - 1);
    int z1 = min(z0 + 1, GL - 1);

    float ux = 1.f - wx, uy = 1.f - wy, uz = 1.f - wz;
    float w00 = uz * uy, w01 = uz * wy, w10 = wz * uy, w11 = wz * wy;

    // 8 corner weights + corner-major LDS offsets (cell*12 floats)
    float cw[8];
    int   co[8];
    cw[0] = w00 * ux;  co[0] = (z0 * GH * GW + y0 * GW + x0) * GC;
    cw[1] = w00 * wx;  co[1] = (z0 * GH * GW + y0 * GW + x1) * GC;
    cw[2] = w01 * ux;  co[2] = (z0 * GH * GW + y1 * GW + x0) * GC;
    cw[3] = w01 * wx;  co[3] = (z0 * GH * GW + y1 * GW + x1) * GC;
    cw[4] = w10 * ux;  co[4] = (z1 * GH * GW + y0 * GW + x0) * GC;
    cw[5] = w10 * wx;  co[5] = (z1 * GH * GW + y0 * GW + x1) * GC;
    cw[6] = w11 * ux;  co[6] = (z1 * GH * GW + y1 * GW + x0) * GC;
    cw[7] = w11 * wx;  co[7] = (z1 * GH * GW + y1 * GW + x1) * GC;

    float A0 = 0.f, A1 = 0.f, A2 = 0.f, A3 = 0.f, A4 = 0.f, A5 = 0.f;
    float A6 = 0.f, A7 = 0.f, A8 = 0.f, A9 = 0.f, A10 = 0.f, A11 = 0.f;
#pragma unroll
    for (int k = 0; k < 8; ++k) {
      const float4* q = (const float4*)(Bg + co[k]);       // 16B aligned
      float4 q0 = q[0], q1 = q[1], q2 = q[2];              // 3x ds_load_b128
      float w = cw[k];
      A0  = fmaf(w, q0.x, A0);   A1  = fmaf(w, q0.y, A1);
      A2  = fmaf(w, q0.z, A2);   A3  = fmaf(w, q0.w, A3);
      A4  = fmaf(w, q1.x, A4);   A5  = fmaf(w, q1.y, A5);
      A6  = fmaf(w, q1.z, A6);   A7  = fmaf(w, q1.w, A7);
      A8  = fmaf(w, q2.x, A8);   A9  = fmaf(w, q2.y, A9);
      A10 = fmaf(w, q2.z, A10);  A11 = fmaf(w, q2.w, A11);
    }

    // out = A[:, :3] @ rgb + A[:, 3]
    opx[3 * p + 0] = fmaf(A0, r, fmaf(A1, g, fmaf(A2,  b, A3)));
    opx[3 * p + 1] = fmaf(A4, r, fmaf(A5, g, fmaf(A6,  b, A7)));
    opx[3 * p + 2] = fmaf(A8, r, fmaf(A9, g, fmaf(A10, b, A11)));
  }
}

extern "C" void kernel_launch(void* const* d_in, const int* in_sizes, int n_in,
                              void* d_out, int out_size, void* d_ws,
                              size_t ws_size, hipStream_t stream) {
  const float* grid_xy = (const float*)d_in[0];   // (N,1,H,W,2) f32
  const float* rgb     = (const float*)d_in[1];   // (N,1,H,W,3) f32
  const float* grids   = (const float*)d_in[2];   // (N,12,8,16,16) f32
  float* out = (float*)d_out;                     // (N,1,H,W,3) f32

  const int N = in_sizes[2] / GRID_ELEMS;         // 4
  if (N <= 0) return;
  const int P = in_sizes[1] / (3 * N);            // 1080*1920 pixels/image

  dim3 block(1024);                               // 32 waves -> fills a WGP
  dim3 grid(128, (unsigned)N);                    // grid-stride over pixels
  hipLaunchKernelGGL(bgrid_slice_kernel, grid, block, 2 * GRID_BYTES, stream,
                     grid_xy, rgb, grids, out, P);
}